// GCN0_70050916598092
// MI455X (gfx1250) — compile-verified
//
#include <hip/hip_runtime.h>

#define N_NODES   100000
#define N_FEAT    128
#define HIDDEN    64
#define N_CLASSES 40
#define N_EDGES   1600000
#define E_TOT     (N_EDGES + N_NODES)   // edges + self loops

typedef __attribute__((ext_vector_type(2))) float v2f;
typedef __attribute__((ext_vector_type(8))) float v8f;

// ---------------------------------------------------------------- utility
__global__ __launch_bounds__(256) void fill_kernel(float* __restrict__ p, int n, float v) {
    int i = blockIdx.x * 256 + threadIdx.x;
    if (i < n) p[i] = v;
}

// deg[dst] += 1 for every real edge (deg pre-initialized to 1.0 for self loop)
__global__ __launch_bounds__(256) void deg_count(const int* __restrict__ ei, float* __restrict__ deg) {
    int e = blockIdx.x * 256 + threadIdx.x;
    if (e < N_EDGES) unsafeAtomicAdd(&deg[ei[N_EDGES + e]], 1.0f);
}

// deg -> rsqrt(deg) in place
__global__ __launch_bounds__(256) void deg_to_dinv(float* __restrict__ deg) {
    int i = blockIdx.x * 256 + threadIdx.x;
    if (i < N_NODES) {
        float d = deg[i];
        deg[i] = (d > 0.0f) ? rsqrtf(d) : 0.0f;
    }
}

// norm[e] = dinv[src] * dinv[dst], once per edge (shared by both layers)
__global__ __launch_bounds__(256) void edge_norm(const int* __restrict__ ei,
                                                 const float* __restrict__ dinv,
                                                 float* __restrict__ nrm) {
    int e = blockIdx.x * 256 + threadIdx.x;
    if (e >= E_TOT) return;
    int s, d;
    if (e < N_EDGES) { s = ei[e]; d = ei[N_EDGES + e]; }
    else             { s = e - N_EDGES; d = s; }
    nrm[e] = dinv[s] * dinv[d];
}

// ---------------------------------------------------------------- GEMM 1 (WMMA fp32)
// hw[100000x64] = x[100000x128] @ W1[128x64]; one wave per 16-row strip.
__global__ __launch_bounds__(32) void gemm1_wmma(const float* __restrict__ x,
                                                 const float* __restrict__ W,
                                                 float* __restrict__ hw) {
    const int rowBase = blockIdx.x * 16;
    const int lane = threadIdx.x;
    const int m  = lane & 15;          // A row / B col / D col
    const int kb = (lane >> 4) << 1;   // fp32 A 16x4: lanes 0-15 K=0,1; lanes 16-31 K=2,3

    v8f c0 = {}, c1 = {}, c2 = {}, c3 = {};
    const float* xr = x + (size_t)(rowBase + m) * N_FEAT;

    for (int kk = 0; kk < N_FEAT; kk += 4) {
        v2f a;
        a.x = xr[kk + kb];
        a.y = xr[kk + kb + 1];
        const float* w0 = W + (size_t)(kk + kb) * HIDDEN;
        const float* w1 = w0 + HIDDEN;
        v2f b;
        b.x = w0[ 0 + m]; b.y = w1[ 0 + m];
        c0 = __builtin_amdgcn_wmma_f32_16x16x4_f32(false, a, false, b, (short)0, c0, false, false);
        b.x = w0[16 + m]; b.y = w1[16 + m];
        c1 = __builtin_amdgcn_wmma_f32_16x16x4_f32(false, a, false, b, (short)0, c1, false, false);
        b.x = w0[32 + m]; b.y = w1[32 + m];
        c2 = __builtin_amdgcn_wmma_f32_16x16x4_f32(false, a, false, b, (short)0, c2, false, false);
        b.x = w0[48 + m]; b.y = w1[48 + m];
        c3 = __builtin_amdgcn_wmma_f32_16x16x4_f32(false, a, false, b, (short)0, c3, false, false);
    }

    // C/D layout: VGPR i -> row (lane<16 ? i : i+8), col = lane&15
    const int rowOff = (lane < 16) ? 0 : 8;
#pragma unroll
    for (int i = 0; i < 8; ++i) {
        float* dst = hw + (size_t)(rowBase + rowOff + i) * HIDDEN;
        dst[ 0 + m] = c0[i];
        dst[16 + m] = c1[i];
        dst[32 + m] = c2[i];
        dst[48 + m] = c3[i];
    }
}

// ---------------------------------------------------------------- GEMM 2 (WMMA fp32)
// h2w[100000x40] = h1[100000x64] @ W2[64x40]; 3 N-tiles, cols >= 40 masked.
__global__ __launch_bounds__(32) void gemm2_wmma(const float* __restrict__ h1,
                                                 const float* __restrict__ W,
                                                 float* __restrict__ h2w) {
    const int rowBase = blockIdx.x * 16;
    const int lane = threadIdx.x;
    const int m  = lane & 15;
    const int kb = (lane >> 4) << 1;

    v8f c0 = {}, c1 = {}, c2 = {};
    const float* hr = h1 + (size_t)(rowBase + m) * HIDDEN;

    for (int kk = 0; kk < HIDDEN; kk += 4) {
        v2f a;
        a.x = hr[kk + kb];
        a.y = hr[kk + kb + 1];
        const float* w0 = W + (size_t)(kk + kb) * N_CLASSES;
        const float* w1 = w0 + N_CLASSES;
        v2f b;
        b.x = w0[ 0 + m]; b.y = w1[ 0 + m];
        c0 = __builtin_amdgcn_wmma_f32_16x16x4_f32(false, a, false, b, (short)0, c0, false, false);
        b.x = w0[16 + m]; b.y = w1[16 + m];
        c1 = __builtin_amdgcn_wmma_f32_16x16x4_f32(false, a, false, b, (short)0, c1, false, false);
        int col2 = 32 + m;
        b.x = (col2 < N_CLASSES) ? w0[col2] : 0.0f;
        b.y = (col2 < N_CLASSES) ? w1[col2] : 0.0f;
        c2 = __builtin_amdgcn_wmma_f32_16x16x4_f32(false, a, false, b, (short)0, c2, false, false);
    }

    const int rowOff = (lane < 16) ? 0 : 8;
#pragma unroll
    for (int i = 0; i < 8; ++i) {
        float* dst = h2w + (size_t)(rowBase + rowOff + i) * N_CLASSES;
        dst[ 0 + m] = c0[i];
        dst[16 + m] = c1[i];
        if (32 + m < N_CLASSES) dst[32 + m] = c2[i];
    }
}

// ---------------------------------------------------------------- edge scatter, 64 feats
// 64 consecutive threads share one edge -> coalesced gather + coalesced fp32 atomics.
__global__ __launch_bounds__(256) void scatter_f64(const float* __restrict__ hsrc,
                                                   const float* __restrict__ nrm,
                                                   const int* __restrict__ ei,
                                                   float* __restrict__ agg) {
    int idx = blockIdx.x * 256 + threadIdx.x;
    int e = idx >> 6;
    if (e >= E_TOT) return;
    int f = idx & 63;
    int s, d;
    if (e < N_EDGES) { s = ei[e]; d = ei[N_EDGES + e]; }
    else             { s = e - N_EDGES; d = s; }
    unsafeAtomicAdd(&agg[(size_t)d * HIDDEN + f], hsrc[(size_t)s * HIDDEN + f] * nrm[e]);
}

// ---------------------------------------------------------------- edge scatter, 40 feats
__global__ __launch_bounds__(256) void scatter_f40(const float* __restrict__ hsrc,
                                                   const float* __restrict__ nrm,
                                                   const int* __restrict__ ei,
                                                   float* __restrict__ agg) {
    int idx = blockIdx.x * 256 + threadIdx.x;
    if (idx >= E_TOT * N_CLASSES) return;
    int e = idx / N_CLASSES;
    int f = idx - e * N_CLASSES;
    int s, d;
    if (e < N_EDGES) { s = ei[e]; d = ei[N_EDGES + e]; }
    else             { s = e - N_EDGES; d = s; }
    unsafeAtomicAdd(&agg[(size_t)d * N_CLASSES + f], hsrc[(size_t)s * N_CLASSES + f] * nrm[e]);
}

// ---------------------------------------------------------------- epilogues
__global__ __launch_bounds__(256) void bias_relu64(float* __restrict__ agg,
                                                   const float* __restrict__ b) {
    int idx = blockIdx.x * 256 + threadIdx.x;
    if (idx < N_NODES * HIDDEN) {
        int f = idx & 63;
        float v = agg[idx] + b[f];
        agg[idx] = fmaxf(v, 0.0f);
    }
}

__global__ __launch_bounds__(256) void bias_out40(const float* __restrict__ agg,
                                                  const float* __restrict__ b,
                                                  float* __restrict__ out) {
    int idx = blockIdx.x * 256 + threadIdx.x;
    if (idx < N_NODES * N_CLASSES) {
        int f = idx % N_CLASSES;
        out[idx] = agg[idx] + b[f];
    }
}

// ---------------------------------------------------------------- launch
extern "C" void kernel_launch(void* const* d_in, const int* in_sizes, int n_in,
                              void* d_out, int out_size, void* d_ws, size_t ws_size,
                              hipStream_t stream) {
    const float* x  = (const float*)d_in[0];   // [100000,128]
    const int*   ei = (const int*)  d_in[1];   // [2,1600000] flattened: row0=src, row1=dst
    const float* W1 = (const float*)d_in[2];   // [128,64]
    const float* b1 = (const float*)d_in[3];   // [64]
    const float* W2 = (const float*)d_in[4];   // [64,40]
    const float* b2 = (const float*)d_in[5];   // [40]
    float* out = (float*)d_out;                // [100000,40]

    // workspace layout (floats)
    float* w    = (float*)d_ws;
    float* dinv = w;                                     // N            (deg -> rsqrt in place)
    float* nrm  = dinv + N_NODES;                        // E_TOT
    float* hbuf = nrm  + E_TOT;                          // N*64 (hw, reused as h2w)
    float* agg1 = hbuf + (size_t)N_NODES * HIDDEN;       // N*64
    float* agg2 = agg1 + (size_t)N_NODES * HIDDEN;       // N*40

    auto cdiv = [](int a, int b) { return (a + b - 1) / b; };

    // degrees (self loop baked in as 1.0) -> dinv -> per-edge norm
    fill_kernel<<<cdiv(N_NODES, 256), 256, 0, stream>>>(dinv, N_NODES, 1.0f);
    deg_count  <<<cdiv(N_EDGES, 256), 256, 0, stream>>>(ei, dinv);
    deg_to_dinv<<<cdiv(N_NODES, 256), 256, 0, stream>>>(dinv);
    edge_norm  <<<cdiv(E_TOT,   256), 256, 0, stream>>>(ei, dinv, nrm);

    // zero both aggregation buffers (contiguous)
    fill_kernel<<<cdiv(N_NODES * (HIDDEN + N_CLASSES), 256), 256, 0, stream>>>(
        agg1, N_NODES * (HIDDEN + N_CLASSES), 0.0f);

    // layer 1: GEMM -> scatter -> bias+relu
    gemm1_wmma <<<N_NODES / 16, 32, 0, stream>>>(x, W1, hbuf);
    scatter_f64<<<cdiv(E_TOT * HIDDEN, 256), 256, 0, stream>>>(hbuf, nrm, ei, agg1);
    bias_relu64<<<cdiv(N_NODES * HIDDEN, 256), 256, 0, stream>>>(agg1, b1);

    // layer 2: GEMM -> scatter -> bias
    gemm2_wmma <<<N_NODES / 16, 32, 0, stream>>>(agg1, W2, hbuf);
    scatter_f40<<<cdiv(E_TOT * N_CLASSES, 256), 256, 0, stream>>>(hbuf, nrm, ei, agg2);
    bias_out40 <<<cdiv(N_NODES * N_CLASSES, 256), 256, 0, stream>>>(agg2, b2, out);
}